// CHESHIRE_53008486367318
// MI455X (gfx1250) — compile-verified
//
#include <hip/hip_runtime.h>
#include <hip/hip_bf16.h>

// ---- problem constants (match reference) ----
#define NN      50000
#define EE      800000
#define IN_DIM  128
#define EMB     64
#define CONV    64
#define GG      512
#define OUTD    128   // 2*CONV

typedef float v2f __attribute__((ext_vector_type(2)));
typedef float v8f __attribute__((ext_vector_type(8)));

__device__ __forceinline__ float clip1(float v) { return fminf(1.0f, fmaxf(-1.0f, v)); }

// =====================================================================
// 1) Encoder: x = clip(feature @ W_enc + b_enc) ; feature [N,128], W [128,64]
//    One wave computes a 16x64 output tile with V_WMMA_F32_16X16X4_F32.
// =====================================================================
__global__ __launch_bounds__(128) void encoder_kernel(
    const float* __restrict__ A, const float* __restrict__ W,
    const float* __restrict__ bias, float* __restrict__ out, int nrows) {
  int wave = (blockIdx.x * blockDim.x + threadIdx.x) >> 5;
  int lane = threadIdx.x & 31;
  int r0 = wave * 16;
  if (r0 >= nrows) return;              // uniform per wave (EXEC stays all-1s)
  int half = lane >> 4, l = lane & 15;

  v8f acc0 = {}, acc1 = {}, acc2 = {}, acc3 = {};
  const float* arow = A + (size_t)(r0 + l) * IN_DIM + 2 * half;

  #pragma unroll 4
  for (int k = 0; k < IN_DIM; k += 4) {
    v2f a; a.x = arow[k]; a.y = arow[k + 1];
    const float* wp = W + (size_t)(k + 2 * half) * EMB + l;
    v2f b0; b0.x = wp[0];  b0.y = wp[EMB + 0];
    v2f b1; b1.x = wp[16]; b1.y = wp[EMB + 16];
    v2f b2; b2.x = wp[32]; b2.y = wp[EMB + 32];
    v2f b3; b3.x = wp[48]; b3.y = wp[EMB + 48];
    acc0 = __builtin_amdgcn_wmma_f32_16x16x4_f32(false, a, false, b0, (short)0, acc0, false, false);
    acc1 = __builtin_amdgcn_wmma_f32_16x16x4_f32(false, a, false, b1, (short)0, acc1, false, false);
    acc2 = __builtin_amdgcn_wmma_f32_16x16x4_f32(false, a, false, b2, (short)0, acc2, false, false);
    acc3 = __builtin_amdgcn_wmma_f32_16x16x4_f32(false, a, false, b3, (short)0, acc3, false, false);
  }
  float bl0 = bias[l], bl1 = bias[16 + l], bl2 = bias[32 + l], bl3 = bias[48 + l];
  #pragma unroll
  for (int j = 0; j < 8; ++j) {
    int row = r0 + j + 8 * half;
    float* o = out + (size_t)row * EMB + l;
    o[0]  = clip1(acc0[j] + bl0);
    o[16] = clip1(acc1[j] + bl1);
    o[32] = clip1(acc2[j] + bl2);
    o[48] = clip1(acc3[j] + bl3);
  }
}

// =====================================================================
// 2) GraphNorm: one block per graph; binary-search segment bounds in sorted
//    batch; 3 passes (mean, var, apply). Stores cnt/start/end for pooling.
// =====================================================================
__device__ __forceinline__ int lower_bound_i(const int* __restrict__ a, int n, int val) {
  int lo = 0, hi = n;
  while (lo < hi) { int mid = (lo + hi) >> 1; if (a[mid] < val) lo = mid + 1; else hi = mid; }
  return lo;
}

__global__ __launch_bounds__(256) void graphnorm_kernel(
    float* __restrict__ x, const int* __restrict__ batch,
    const float* __restrict__ gnw, const float* __restrict__ gnb,
    const float* __restrict__ gms, float* __restrict__ cnt_out,
    int* __restrict__ gstart, int* __restrict__ gend, int n) {
  int g = blockIdx.x;
  int tid = threadIdx.x;
  int start = lower_bound_i(batch, n, g);
  int end   = lower_bound_i(batch, n, g + 1);
  float cntf = fmaxf((float)(end - start), 1.0f);
  if (tid == 0) { cnt_out[g] = cntf; gstart[g] = start; gend[g] = end; }

  __shared__ float sbuf[256];
  __shared__ float smean[EMB];
  __shared__ float srstd[EMB];
  int f = tid & 63, r = tid >> 6;

  // pass 1: mean
  float p = 0.0f;
  for (int i = start + r; i < end; i += 4) p += x[(size_t)i * EMB + f];
  sbuf[tid] = p;
  __syncthreads();
  if (tid < EMB)
    smean[tid] = (sbuf[tid] + sbuf[tid + 64] + sbuf[tid + 128] + sbuf[tid + 192]) / cntf;
  __syncthreads();

  float mscaled = smean[f] * gms[f];

  // pass 2: variance of centered
  float q = 0.0f;
  for (int i = start + r; i < end; i += 4) {
    float c = x[(size_t)i * EMB + f] - mscaled;
    q += c * c;
  }
  sbuf[tid] = q;
  __syncthreads();
  if (tid < EMB) {
    float var = (sbuf[tid] + sbuf[tid + 64] + sbuf[tid + 128] + sbuf[tid + 192]) / cntf;
    srstd[tid] = 1.0f / sqrtf(var + 1e-5f);
  }
  __syncthreads();

  // pass 3: apply (in place)
  float w = gnw[f], b = gnb[f], rs = srstd[f];
  for (int i = start + r; i < end; i += 4) {
    size_t idx = (size_t)i * EMB + f;
    x[idx] = w * (x[idx] - mscaled) * rs + b;
  }
}

// =====================================================================
// 3) Degree / dinv
// =====================================================================
__global__ void deg_kernel(const int* __restrict__ src, float* __restrict__ deg, int e) {
  int i = blockIdx.x * blockDim.x + threadIdx.x;
  if (i < e) atomicAdd(&deg[src[i]], 1.0f);
}

__global__ void dinv_kernel(const float* __restrict__ deg, float* __restrict__ dinv, int n) {
  int i = blockIdx.x * blockDim.x + threadIdx.x;
  if (i < n) {
    float d = deg[i];
    dinv[i] = (d > 0.0f) ? (1.0f / sqrtf(fmaxf(d, 1.0f))) : 0.0f;
  }
}

// =====================================================================
// 4) Sparse propagation: out[dst] += (-dinv[src]*dinv[dst]) * in[src]
//    64 lanes per edge -> coalesced gather + coalesced f32 atomic scatter
//    (node matrix is 12.8 MB: fully L2-resident on MI455X's 192 MB L2).
// =====================================================================
__global__ __launch_bounds__(256) void prop_kernel(
    const float* __restrict__ in, float* __restrict__ out,
    const int* __restrict__ src, const int* __restrict__ dst,
    const float* __restrict__ dinv, int e) {
  int eid = blockIdx.x * 4 + (threadIdx.x >> 6);
  if (eid >= e) return;
  int f = threadIdx.x & 63;
  int s = src[eid], d = dst[eid];
  float w = -dinv[s] * dinv[d];
  atomicAdd(&out[(size_t)d * EMB + f], w * in[(size_t)s * EMB + f]);
}

// Tx2 = 2*prop(Tx1) - Tx0  (in place on the prop result)
__global__ void tx2_kernel(float* __restrict__ p2, const float* __restrict__ x0, int n) {
  int i = blockIdx.x * blockDim.x + threadIdx.x;
  if (i < n) p2[i] = 2.0f * p2[i] - x0[i];
}

// =====================================================================
// 5) ChebConv combine: h = clip(T0@W0 + T1@W1 + T2@W2 + b), all 64x64 GEMMs
// =====================================================================
__global__ __launch_bounds__(128) void cheb_kernel(
    const float* __restrict__ T0, const float* __restrict__ T1,
    const float* __restrict__ T2, const float* __restrict__ Wc,
    const float* __restrict__ bias, float* __restrict__ out, int nrows) {
  int wave = (blockIdx.x * blockDim.x + threadIdx.x) >> 5;
  int lane = threadIdx.x & 31;
  int r0 = wave * 16;
  if (r0 >= nrows) return;
  int half = lane >> 4, l = lane & 15;

  v8f acc0 = {}, acc1 = {}, acc2 = {}, acc3 = {};

  auto accumulate = [&](const float* __restrict__ M, const float* __restrict__ Wm) {
    const float* arow = M + (size_t)(r0 + l) * EMB + 2 * half;
    #pragma unroll 4
    for (int k = 0; k < EMB; k += 4) {
      v2f a; a.x = arow[k]; a.y = arow[k + 1];
      const float* wp = Wm + (size_t)(k + 2 * half) * CONV + l;
      v2f b0; b0.x = wp[0];  b0.y = wp[CONV + 0];
      v2f b1; b1.x = wp[16]; b1.y = wp[CONV + 16];
      v2f b2; b2.x = wp[32]; b2.y = wp[CONV + 32];
      v2f b3; b3.x = wp[48]; b3.y = wp[CONV + 48];
      acc0 = __builtin_amdgcn_wmma_f32_16x16x4_f32(false, a, false, b0, (short)0, acc0, false, false);
      acc1 = __builtin_amdgcn_wmma_f32_16x16x4_f32(false, a, false, b1, (short)0, acc1, false, false);
      acc2 = __builtin_amdgcn_wmma_f32_16x16x4_f32(false, a, false, b2, (short)0, acc2, false, false);
      acc3 = __builtin_amdgcn_wmma_f32_16x16x4_f32(false, a, false, b3, (short)0, acc3, false, false);
    }
  };
  accumulate(T0, Wc);
  accumulate(T1, Wc + EMB * CONV);
  accumulate(T2, Wc + 2 * EMB * CONV);

  float bl0 = bias[l], bl1 = bias[16 + l], bl2 = bias[32 + l], bl3 = bias[48 + l];
  #pragma unroll
  for (int j = 0; j < 8; ++j) {
    int row = r0 + j + 8 * half;
    float* o = out + (size_t)row * CONV + l;
    o[0]  = clip1(acc0[j] + bl0);
    o[16] = clip1(acc1[j] + bl1);
    o[32] = clip1(acc2[j] + bl2);
    o[48] = clip1(acc3[j] + bl3);
  }
}

// =====================================================================
// 6) Pooling: per-graph (max-min) and RMS -> y [G, 128]
// =====================================================================
__global__ __launch_bounds__(256) void pool_kernel(
    const float* __restrict__ h, const int* __restrict__ gstart,
    const int* __restrict__ gend, const float* __restrict__ cnt,
    float* __restrict__ y) {
  int g = blockIdx.x;
  int tid = threadIdx.x;
  int start = gstart[g], end = gend[g];
  int f = tid & 63, r = tid >> 6;

  float mx = -3.402823466e38f, mn = 3.402823466e38f, sq = 0.0f;
  for (int i = start + r; i < end; i += 4) {
    float v = h[(size_t)i * CONV + f];
    mx = fmaxf(mx, v); mn = fminf(mn, v); sq += v * v;
  }
  __shared__ float smx[256], smn[256], ssq[256];
  smx[tid] = mx; smn[tid] = mn; ssq[tid] = sq;
  __syncthreads();
  if (tid < 64) {
    mx = fmaxf(fmaxf(smx[tid], smx[tid + 64]), fmaxf(smx[tid + 128], smx[tid + 192]));
    mn = fminf(fminf(smn[tid], smn[tid + 64]), fminf(smn[tid + 128], smn[tid + 192]));
    sq = ssq[tid] + ssq[tid + 64] + ssq[tid + 128] + ssq[tid + 192];
    if (end <= start) { mx = 0.0f; mn = 0.0f; }
    y[(size_t)g * OUTD + tid]      = mx - mn;
    y[(size_t)g * OUTD + 64 + tid] = sqrtf(sq / cnt[g]);
  }
}

// =====================================================================
// 7) Output linear: out = y @ W_out + b_out  ([512,128] @ [128,128])
// =====================================================================
__global__ __launch_bounds__(128) void out_kernel(
    const float* __restrict__ Y, const float* __restrict__ W,
    const float* __restrict__ bias, float* __restrict__ out, int nrows) {
  int wave = (blockIdx.x * blockDim.x + threadIdx.x) >> 5;
  int lane = threadIdx.x & 31;
  int r0 = wave * 16;
  if (r0 >= nrows) return;
  int half = lane >> 4, l = lane & 15;

  v8f acc[8];
  #pragma unroll
  for (int t = 0; t < 8; ++t) acc[t] = (v8f){};

  const float* arow = Y + (size_t)(r0 + l) * OUTD + 2 * half;
  for (int k = 0; k < OUTD; k += 4) {
    v2f a; a.x = arow[k]; a.y = arow[k + 1];
    const float* wp = W + (size_t)(k + 2 * half) * OUTD + l;
    #pragma unroll
    for (int t = 0; t < 8; ++t) {
      v2f b; b.x = wp[16 * t]; b.y = wp[OUTD + 16 * t];
      acc[t] = __builtin_amdgcn_wmma_f32_16x16x4_f32(false, a, false, b, (short)0, acc[t], false, false);
    }
  }
  #pragma unroll
  for (int t = 0; t < 8; ++t) {
    float bl = bias[16 * t + l];
    #pragma unroll
    for (int j = 0; j < 8; ++j) {
      int row = r0 + j + 8 * half;
      out[(size_t)row * OUTD + 16 * t + l] = acc[t][j] + bl;
    }
  }
}

// =====================================================================
extern "C" void kernel_launch(void* const* d_in, const int* in_sizes, int n_in,
                              void* d_out, int out_size, void* d_ws, size_t ws_size,
                              hipStream_t stream) {
  (void)in_sizes; (void)n_in; (void)out_size; (void)ws_size;

  const float* feature = (const float*)d_in[0];
  const float* W_enc   = (const float*)d_in[1];
  const float* b_enc   = (const float*)d_in[2];
  const float* gn_w    = (const float*)d_in[3];
  const float* gn_b    = (const float*)d_in[4];
  const float* gn_ms   = (const float*)d_in[5];
  const float* cheb_W  = (const float*)d_in[6];
  const float* cheb_b  = (const float*)d_in[7];
  const float* W_out   = (const float*)d_in[8];
  const float* b_out   = (const float*)d_in[9];
  const int*   eidx    = (const int*)d_in[10];
  const int*   batch   = (const int*)d_in[11];
  const int* src = eidx;
  const int* dst = eidx + EE;

  // workspace carve-out (~52 MB total)
  char* ws = (char*)d_ws;
  size_t off = 0;
  auto carve = [&](size_t bytes) -> void* {
    void* p = ws + off;
    off = (off + bytes + 255) & ~(size_t)255;
    return p;
  };
  float* x    = (float*)carve((size_t)NN * EMB * 4);   // Tx0 (normalized)
  float* tx1  = (float*)carve((size_t)NN * EMB * 4);
  float* tx2  = (float*)carve((size_t)NN * EMB * 4);   // holds prop(Tx1), then Tx2
  float* h    = (float*)carve((size_t)NN * CONV * 4);
  float* deg  = (float*)carve((size_t)NN * 4);
  float* dinv = (float*)carve((size_t)NN * 4);
  float* cnt  = (float*)carve((size_t)GG * 4);
  int*   gst  = (int*)carve((size_t)GG * 4);
  int*   gen  = (int*)carve((size_t)GG * 4);
  float* ypool = (float*)carve((size_t)GG * OUTD * 4);

  hipMemsetAsync(tx1, 0, (size_t)NN * EMB * 4, stream);
  hipMemsetAsync(tx2, 0, (size_t)NN * EMB * 4, stream);
  hipMemsetAsync(deg, 0, (size_t)NN * 4, stream);

  // 1) encoder GEMM (WMMA f32): 3125 row-tiles, 4 waves/block
  int row_tiles = (NN + 15) / 16;                      // 3125
  int enc_blocks = (row_tiles + 3) / 4;                // 782
  encoder_kernel<<<enc_blocks, 128, 0, stream>>>(feature, W_enc, b_enc, x, NN);

  // 2) GraphNorm (in place on x)
  graphnorm_kernel<<<GG, 256, 0, stream>>>(x, batch, gn_w, gn_b, gn_ms, cnt, gst, gen, NN);

  // 3) degrees
  deg_kernel<<<(EE + 255) / 256, 256, 0, stream>>>(src, deg, EE);
  dinv_kernel<<<(NN + 255) / 256, 256, 0, stream>>>(deg, dinv, NN);

  // 4) Tx1 = L_hat @ x ; Tx2 = 2*L_hat @ Tx1 - x
  prop_kernel<<<(EE + 3) / 4, 256, 0, stream>>>(x, tx1, src, dst, dinv, EE);
  prop_kernel<<<(EE + 3) / 4, 256, 0, stream>>>(tx1, tx2, src, dst, dinv, EE);
  tx2_kernel<<<((NN * EMB) + 255) / 256, 256, 0, stream>>>(tx2, x, NN * EMB);

  // 5) Cheb combine GEMM (WMMA f32)
  cheb_kernel<<<enc_blocks, 128, 0, stream>>>(x, tx1, tx2, cheb_W, cheb_b, h, NN);

  // 6) pooling
  pool_kernel<<<GG, 256, 0, stream>>>(h, gst, gen, cnt, ypool);

  // 7) output GEMM (WMMA f32): 32 row-tiles, 4 waves/block
  out_kernel<<<8, 128, 0, stream>>>(ypool, W_out, b_out, (float*)d_out, GG);
}